// CloudRasterizer_49675591745755
// MI455X (gfx1250) — compile-verified
//
#include <hip/hip_runtime.h>
#include <hip/hip_bf16.h>

// CloudRasterizer for MI455X (gfx1250, wave32).
// - 64 MB f32 cube stays resident in the 192 MB L2; all 32M scatter-adds are
//   native global_atomic_add_f32 (no-return) resolving at L2.
// - 64 MB of streamed point data is loaded with TH=NT so it does not evict
//   the cube from L2.
// - 2 points per thread: b128/b64 streaming loads, 16 independent atomics in
//   flight per lane for VMEM pipelining.

#define NPIX      512
#define NVCH      64
#define PIXSCALE  0.1f
#define DVCH      10.0f
#define FOV_HALF  25.55f   // 0.5*(512-1)*0.1

typedef float  v2f __attribute__((ext_vector_type(2)));
typedef float  v4f __attribute__((ext_vector_type(4)));

__global__ void __launch_bounds__(256)
cr_zero_cube(v4f* __restrict__ out, int n4) {
    int i = blockIdx.x * blockDim.x + threadIdx.x;
    if (i < n4) {
        v4f z = {0.0f, 0.0f, 0.0f, 0.0f};
        out[i] = z;
    }
}

__device__ __forceinline__ void deposit(float* __restrict__ cube,
                                        float ra, float dec, float v, float f) {
    float x = (ra  + FOV_HALF) / PIXSCALE;
    float y = (dec + FOV_HALF) / PIXSCALE;
    float w = v / DVCH;                        // VEL0 == 0

    float fx0 = floorf(x), fy0 = floorf(y), fw0 = floorf(w);
    int ix0 = (int)fx0, iy0 = (int)fy0, iv0 = (int)fw0;
    float fx = x - fx0, fy = y - fy0, fv = w - fw0;

    bool ok = (ix0 >= 0) && (ix0 < NPIX - 1) &&
              (iy0 >= 0) && (iy0 < NPIX - 1) &&
              (iv0 >= 0) && (iv0 < NVCH - 1);
    if (!ok) f = 0.0f;                         // zero flux + clipped index == ref
    ix0 = min(max(ix0, 0), NPIX - 2);
    iy0 = min(max(iy0, 0), NPIX - 2);
    iv0 = min(max(iv0, 0), NVCH - 2);

    // wy uses the (0,1,0,1,...) selector (same pattern as wv) per the
    // reference quirk => only two distinct y*v products:
    float omx = 1.0f - fx;
    float A   = (1.0f - fy) * (1.0f - fv);     // corners with dv=0
    float B   = fy * fv;                       // corners with dv=1
    float w0A = f * omx * A, w1A = f * fx * A;
    float w0B = f * omx * B, w1B = f * fx * B;

    const int SY = NPIX;
    const int SV = NPIX * NPIX;
    float* b = cube + ((iv0 * NPIX + iy0) * NPIX + ix0);

    // 8 corners -> 8 native fp32 global atomics (resolve in L2)
    unsafeAtomicAdd(b,               w0A);
    unsafeAtomicAdd(b + 1,           w1A);
    unsafeAtomicAdd(b + SY,          w0A);
    unsafeAtomicAdd(b + SY + 1,      w1A);
    unsafeAtomicAdd(b + SV,          w0B);
    unsafeAtomicAdd(b + SV + 1,      w1B);
    unsafeAtomicAdd(b + SV + SY,     w0B);
    unsafeAtomicAdd(b + SV + SY + 1, w1B);
}

__global__ void __launch_bounds__(256)
cr_scatter_x2(const float* __restrict__ pos,   // [M,2] (ra, dec)
              const float* __restrict__ vel,   // [M]
              const float* __restrict__ flx,   // [M]
              float* __restrict__ cube,        // [64*512*512]
              int M) {
    int i  = blockIdx.x * blockDim.x + threadIdx.x;
    int j0 = 2 * i;
    if (j0 >= M) return;

    if (j0 + 1 < M) {
        // vectorized path: two points, b128 + 2x b64 non-temporal loads
        v4f p  = __builtin_nontemporal_load((const v4f*)pos + i);
        v2f vv = __builtin_nontemporal_load((const v2f*)vel + i);
        v2f ff = __builtin_nontemporal_load((const v2f*)flx + i);
        deposit(cube, p.x, p.y, vv.x, ff.x);
        deposit(cube, p.z, p.w, vv.y, ff.y);
    } else {
        // odd tail point
        v2f p    = __builtin_nontemporal_load((const v2f*)pos + j0);
        float vv = __builtin_nontemporal_load(vel + j0);
        float ff = __builtin_nontemporal_load(flx + j0);
        deposit(cube, p.x, p.y, vv, ff);
    }
}

extern "C" void kernel_launch(void* const* d_in, const int* in_sizes, int n_in,
                              void* d_out, int out_size, void* d_ws, size_t ws_size,
                              hipStream_t stream) {
    const float* pos = (const float*)d_in[0];  // pos_img [N,S,2] f32
    const float* vel = (const float*)d_in[1];  // vel_chan [N,S]
    const float* flx = (const float*)d_in[2];  // flux     [N,S]
    float* cube = (float*)d_out;               // [64,512,512] f32

    int M = in_sizes[1];                       // N*S = 4,000,000 points

    // 1) zero the cube (harness poisons d_out; re-init every call)
    int n4 = out_size / 4;                     // 128-bit stores
    int zb = (n4 + 255) / 256;
    cr_zero_cube<<<zb, 256, 0, stream>>>((v4f*)cube, n4);

    // 2) scatter-add, two points per thread
    int T  = (M + 1) / 2;
    int sb = (T + 255) / 256;
    cr_scatter_x2<<<sb, 256, 0, stream>>>(pos, vel, flx, cube, M);
}